// MultiHeadAttentionModule_20392504721972
// MI455X (gfx1250) — compile-verified
//
#include <hip/hip_runtime.h>
#include <math.h>

typedef float v2f __attribute__((ext_vector_type(2)));
typedef float v8f __attribute__((ext_vector_type(8)));

#define C_CH   64
#define NHEAD  8
#define HDIM   8
#define NN     6144          // 8*24*32
#define NQT    (NN / 16)     // 384 tiles of 16

__device__ __forceinline__ v8f wmma4(v2f a, v2f b, v8f c) {
    // D(16x16,f32) = A(16x4,f32) * B(4x16,f32) + C
    return __builtin_amdgcn_wmma_f32_16x16x4_f32(
        /*neg_a=*/false, a, /*neg_b=*/false, b,
        /*c_mod=*/(short)0, c, /*reuse_a=*/false, /*reuse_b=*/false);
}

__device__ __forceinline__ float shx16(float v) {
    return __shfl_xor(v, 16, 32);
}

// ---------------------------------------------------------------------------
// Kernel 1: fused 1x1x1-conv projections  q/k/v = W * x + b
// x: [64][N] fp32.  Output qkv: [3][64][N]  (channel = head*8 + dim)
// One wave computes one 16(o) x 16(n) tile with 16 chained f32 WMMAs (K=64).
// ---------------------------------------------------------------------------
__global__ __launch_bounds__(32) void proj_wmma_kernel(
    const float* __restrict__ x,
    const float* __restrict__ wq, const float* __restrict__ bq,
    const float* __restrict__ wk, const float* __restrict__ bk,
    const float* __restrict__ wv, const float* __restrict__ bv,
    float* __restrict__ qkv)
{
    const int lane = threadIdx.x;
    const int l16  = lane & 15;
    const int hs   = (lane >> 4) * 2;   // K-half offset per A/B f32 layout

    int t   = blockIdx.x;
    const int nt = t % NQT;             // n tile
    const int ot = (t / NQT) & 3;       // o tile (64/16)
    const int m  = t / (NQT * 4);       // 0=q 1=k 2=v

    const float* w    = (m == 0) ? wq : (m == 1) ? wk : wv;
    const float* bias = (m == 0) ? bq : (m == 1) ? bk : bv;
    float* out = qkv + (size_t)m * C_CH * NN;

    const int o_row = ot * 16 + l16;    // A: M = lane%16
    const int n_col = nt * 16 + l16;    // B: N = lane%16

    v8f acc = {};
    #pragma unroll
    for (int c0 = 0; c0 < C_CH; c0 += 4) {
        v2f a, b;
        a.x = w[o_row * C_CH + c0 + hs];
        a.y = w[o_row * C_CH + c0 + hs + 1];
        b.x = x[(size_t)(c0 + hs)     * NN + n_col];
        b.y = x[(size_t)(c0 + hs + 1) * NN + n_col];
        acc = wmma4(a, b, acc);
    }

    // D layout: vgpr r -> row ot*16 + r (+8 for upper half lanes), col = lane%16
    const int row_base = ot * 16 + ((lane < 16) ? 0 : 8);
    #pragma unroll
    for (int r = 0; r < 8; ++r) {
        int o = row_base + r;
        out[(size_t)o * NN + n_col] = acc[r] + bias[o];
    }
}

// ---------------------------------------------------------------------------
// Kernel 2: flash attention, one wave per (head, 16-query tile).
// qkv layout [3][head][dim][N].  Energy computed transposed: E^T[kv][q] so the
// softmax (over keys) reduces within-lane over 8 accumulator VGPRs + one
// shfl_xor(16).  O^T[dim][q] accumulated with V^T as the A matrix (dim padded
// 8->16 with zeros).  Final: out = gamma * O/l + x.
// ---------------------------------------------------------------------------
__global__ __launch_bounds__(32) void flash_attn_kernel(
    const float* __restrict__ x,
    const float* __restrict__ qkv,
    const float* __restrict__ gamma,
    float* __restrict__ out)
{
    const int lane = threadIdx.x;
    const int l16  = lane & 15;
    const int hs   = (lane >> 4) * 2;

    const int qt   = blockIdx.x % NQT;
    const int head = blockIdx.x / NQT;

    const float* qh = qkv + (size_t)head * HDIM * NN;
    const float* kh = qkv + (size_t)C_CH * NN     + (size_t)head * HDIM * NN;
    const float* vh = qkv + (size_t)2 * C_CH * NN + (size_t)head * HDIM * NN;

    const int qcol = qt * 16 + l16;

    // B operand (Q^T, 8x16) for the two K=4 steps -- invariant over key loop.
    v2f bq0, bq1;
    bq0.x = qh[(size_t)(hs + 0) * NN + qcol];
    bq0.y = qh[(size_t)(hs + 1) * NN + qcol];
    bq1.x = qh[(size_t)(hs + 4) * NN + qcol];
    bq1.y = qh[(size_t)(hs + 5) * NN + qcol];

    // V^T row for this lane's dim (pad dims 8..15 with zeros).
    const int dim = l16;
    const bool dim_ok = (dim < HDIM);
    const float* vrow = vh + (size_t)(dim_ok ? dim : (HDIM - 1)) * NN;

    const float scale = 0.35355339059327373f;   // 1/sqrt(head_dim)

    float m_run = -INFINITY;
    float l_run = 0.0f;
    v8f o_acc = {};

    for (int kt = 0; kt < NQT; ++kt) {
        const int kvrow = kt * 16 + l16;

        // A operand: K tile rows (M = kv, K = dim)
        v2f ak0, ak1;
        ak0.x = kh[(size_t)(hs + 0) * NN + kvrow];
        ak0.y = kh[(size_t)(hs + 1) * NN + kvrow];
        ak1.x = kh[(size_t)(hs + 4) * NN + kvrow];
        ak1.y = kh[(size_t)(hs + 5) * NN + kvrow];

        v8f e = {};
        e = wmma4(ak0, bq0, e);
        e = wmma4(ak1, bq1, e);      // E^T[kv][q], rows in-lane across 8 vgprs

        // ---- online softmax over keys (per query column == per lane) ----
        float tmax = e[0];
        #pragma unroll
        for (int r = 1; r < 8; ++r) tmax = fmaxf(tmax, e[r]);
        tmax = fmaxf(tmax, shx16(tmax));
        tmax *= scale;

        const float m_new = fmaxf(m_run, tmax);
        const float corr  = __expf(m_run - m_new);

        v8f p;
        float psum = 0.0f;
        #pragma unroll
        for (int r = 0; r < 8; ++r) {
            p[r] = __expf(e[r] * scale - m_new);
            psum += p[r];
        }
        psum += shx16(psum);

        l_run = l_run * corr + psum;
        m_run = m_new;
        #pragma unroll
        for (int r = 0; r < 8; ++r) o_acc[r] *= corr;

        // ---- O^T += V^T * P : 4 K=4 steps; P re-shaped D->B via shfl/select
        const bool lo = (lane < 16);
        const int kb  = kt * 16;
        #pragma unroll
        for (int kk = 0; kk < 16; kk += 4) {
            v2f bp, av;
            if (kk < 8) {   // kv rows kk..kk+3 live in lower-half lanes
                bp.x = lo ? p[kk + 0] : shx16(p[kk + 2]);
                bp.y = lo ? p[kk + 1] : shx16(p[kk + 3]);
            } else {        // kv rows kk..kk+3 live in upper-half lanes
                bp.x = lo ? shx16(p[kk - 8 + 0]) : p[kk - 8 + 2];
                bp.y = lo ? shx16(p[kk - 8 + 1]) : p[kk - 8 + 3];
            }
            const int kv0 = kb + kk + hs;
            av.x = dim_ok ? vrow[kv0 + 0] : 0.0f;
            av.y = dim_ok ? vrow[kv0 + 1] : 0.0f;
            o_acc = wmma4(av, bp, o_acc);
        }
    }

    // ---- epilogue: out[c][n] = gamma * O/l + x,  c = head*8 + dim ----
    const float g   = gamma[0];
    const float inv = 1.0f / l_run;
    if (lane < 16) {                      // lanes 0-15 hold dims 0-7 (vgpr r)
        #pragma unroll
        for (int r = 0; r < 8; ++r) {
            size_t idx = (size_t)(head * HDIM + r) * NN + qcol;
            out[idx] = g * (o_acc[r] * inv) + x[idx];
        }
    }
}

// ---------------------------------------------------------------------------
extern "C" void kernel_launch(void* const* d_in, const int* in_sizes, int n_in,
                              void* d_out, int out_size, void* d_ws, size_t ws_size,
                              hipStream_t stream) {
    const float* x     = (const float*)d_in[0];
    const float* wq    = (const float*)d_in[1];
    const float* bq    = (const float*)d_in[2];
    const float* wk    = (const float*)d_in[3];
    const float* bk    = (const float*)d_in[4];
    const float* wv    = (const float*)d_in[5];
    const float* bv    = (const float*)d_in[6];
    const float* gamma = (const float*)d_in[7];
    float* out = (float*)d_out;
    float* qkv = (float*)d_ws;            // 3 * 64 * 6144 floats = 4.7 MB

    proj_wmma_kernel<<<dim3(3 * 4 * NQT), dim3(32), 0, stream>>>(
        x, wq, bq, wk, bk, wv, bv, qkv);
    flash_attn_kernel<<<dim3(NHEAD * NQT), dim3(32), 0, stream>>>(
        x, qkv, gamma, out);
}